// CSM_62216896250027
// MI455X (gfx1250) — compile-verified
//
#include <hip/hip_runtime.h>
#include <stdint.h>

// Problem constants (match the reference).
#define B_DIM 8192
#define S_DIM 7
#define E_DIM 2048

#define TPB   128          // threads per block
#define VEC   4            // channels per thread (float4)
#define EBLK  (TPB * VEC)  // 512 channels per block
#define GRID_Y 1024        // blocks striding over batch dim

// ---------------- CDNA5-specific primitives ----------------

// gfx1250 hardware tanh (trans pipe) with graceful fallback.
__device__ __forceinline__ float htanh(float x) {
#if __has_builtin(__builtin_amdgcn_tanhf)
    return __builtin_amdgcn_tanhf(x);
#elif __has_builtin(__builtin_amdgcn_tanh_f32)
    return __builtin_amdgcn_tanh_f32(x);
#else
    return tanhf(x);
#endif
}

// Async global -> LDS 128-bit copy (GV mode), tracked by ASYNCcnt.
// vdst = per-lane LDS byte offset, vaddr = 64-bit global address.
__device__ __forceinline__ void async_copy_b128(uint32_t lds_byte_off, const void* gaddr) {
    asm volatile("global_load_async_to_lds_b128 %0, %1, off"
                 :
                 : "v"(lds_byte_off), "v"(gaddr)
                 : "memory");
}

#if __has_builtin(__builtin_amdgcn_s_wait_asynccnt)
#define WAIT_ASYNC(N) __builtin_amdgcn_s_wait_asynccnt(N)
#else
#define WAIT_ASYNC(N) asm volatile("s_wait_asynccnt %0" ::"i"(N) : "memory")
#endif

// ---------------- float4 helpers ----------------

__device__ __forceinline__ float4 f4_mul(float4 a, float4 b) {
    return make_float4(a.x * b.x, a.y * b.y, a.z * b.z, a.w * b.w);
}
__device__ __forceinline__ float4 f4_fma(float4 a, float4 b, float4 c) {
    return make_float4(fmaf(a.x, b.x, c.x), fmaf(a.y, b.y, c.y),
                       fmaf(a.z, b.z, c.z), fmaf(a.w, b.w, c.w));
}
__device__ __forceinline__ float4 f4_tanh(float4 a) {
    return make_float4(htanh(a.x), htanh(a.y), htanh(a.z), htanh(a.w));
}
__device__ __forceinline__ float4 ldg4(const float* p) {
    return *reinterpret_cast<const float4*>(p);
}

// ---------------- kernel ----------------

__global__ __launch_bounds__(TPB) void CSM_conv_tanh_chain(
    const float* __restrict__ X,      // (B, S, E)
    const float* __restrict__ conv1,  // (2, E)
    const float* __restrict__ conv2,  // (2, E)
    const float* __restrict__ conv3,  // (3, E)
    const float* __restrict__ conv4,  // (3, E)
    float* __restrict__ out)          // (B, E)
{
    __shared__ __align__(16) float smem[2][S_DIM][EBLK];

    const int tid = threadIdx.x;
    const int e   = blockIdx.x * EBLK + tid * VEC;   // channel base for this thread

    // Per-channel weights: loaded once, reused across all batches this block handles.
    const float4 k1a = ldg4(conv1 + 0 * E_DIM + e);
    const float4 k1b = ldg4(conv1 + 1 * E_DIM + e);
    const float4 k2a = ldg4(conv2 + 0 * E_DIM + e);
    const float4 k2b = ldg4(conv2 + 1 * E_DIM + e);
    const float4 k3a = ldg4(conv3 + 0 * E_DIM + e);
    const float4 k3b = ldg4(conv3 + 1 * E_DIM + e);
    const float4 k3c = ldg4(conv3 + 2 * E_DIM + e);
    const float4 k4a = ldg4(conv4 + 0 * E_DIM + e);
    const float4 k4b = ldg4(conv4 + 1 * E_DIM + e);
    const float4 k4c = ldg4(conv4 + 2 * E_DIM + e);

    // Thread's LDS byte offset within a buffer row.
    const uint32_t lds_thread = (uint32_t)(uintptr_t)(&smem[0][0][0]) + (uint32_t)(tid * VEC * 4);
    const uint32_t ROW_BYTES  = EBLK * 4;           // 2048 B per staged row
    const uint32_t BUF_BYTES  = S_DIM * ROW_BYTES;  // 14 KB per buffer

    int b = (int)blockIdx.y;
    if (b >= B_DIM) return;

    // Issue the 7 async row copies for one batch into buffer `buf`.
    auto issue = [&](int bb, int buf) {
        const float* gb = X + (size_t)bb * (S_DIM * E_DIM) + e;
        const uint32_t base = lds_thread + (uint32_t)buf * BUF_BYTES;
#pragma unroll
        for (int s = 0; s < S_DIM; ++s)
            async_copy_b128(base + (uint32_t)s * ROW_BYTES, gb + (size_t)s * E_DIM);
    };

    // Prologue: stage first batch.
    issue(b, 0);
    int buf = 0;

    for (; b < B_DIM; b += (int)gridDim.y) {
        const int bn = b + (int)gridDim.y;
        if (bn < B_DIM) {
            issue(bn, buf ^ 1);   // prefetch next batch into other buffer
            WAIT_ASYNC(7);        // drain older group (in-order completion)
        } else {
            WAIT_ASYNC(0);
        }

        // Read this thread's 7 staged values per channel (ds_load_b128).
        float4 x0 = *reinterpret_cast<const float4*>(&smem[buf][0][tid * VEC]);
        float4 x1 = *reinterpret_cast<const float4*>(&smem[buf][1][tid * VEC]);
        float4 x2 = *reinterpret_cast<const float4*>(&smem[buf][2][tid * VEC]);
        float4 x3 = *reinterpret_cast<const float4*>(&smem[buf][3][tid * VEC]);
        float4 x4 = *reinterpret_cast<const float4*>(&smem[buf][4][tid * VEC]);
        float4 x5 = *reinterpret_cast<const float4*>(&smem[buf][5][tid * VEC]);
        float4 x6 = *reinterpret_cast<const float4*>(&smem[buf][6][tid * VEC]);

        // Stage 1: h=2, 7 -> 6
        float4 a0 = f4_tanh(f4_fma(x1, k1b, f4_mul(x0, k1a)));
        float4 a1 = f4_tanh(f4_fma(x2, k1b, f4_mul(x1, k1a)));
        float4 a2 = f4_tanh(f4_fma(x3, k1b, f4_mul(x2, k1a)));
        float4 a3 = f4_tanh(f4_fma(x4, k1b, f4_mul(x3, k1a)));
        float4 a4 = f4_tanh(f4_fma(x5, k1b, f4_mul(x4, k1a)));
        float4 a5 = f4_tanh(f4_fma(x6, k1b, f4_mul(x5, k1a)));

        // Stage 2: h=2, 6 -> 5
        float4 b0 = f4_tanh(f4_fma(a1, k2b, f4_mul(a0, k2a)));
        float4 b1 = f4_tanh(f4_fma(a2, k2b, f4_mul(a1, k2a)));
        float4 b2 = f4_tanh(f4_fma(a3, k2b, f4_mul(a2, k2a)));
        float4 b3 = f4_tanh(f4_fma(a4, k2b, f4_mul(a3, k2a)));
        float4 b4 = f4_tanh(f4_fma(a5, k2b, f4_mul(a4, k2a)));

        // Stage 3: h=3, 5 -> 3
        float4 c0 = f4_tanh(f4_fma(b2, k3c, f4_fma(b1, k3b, f4_mul(b0, k3a))));
        float4 c1 = f4_tanh(f4_fma(b3, k3c, f4_fma(b2, k3b, f4_mul(b1, k3a))));
        float4 c2 = f4_tanh(f4_fma(b4, k3c, f4_fma(b3, k3b, f4_mul(b2, k3a))));

        // Stage 4: h=3, 3 -> 1
        float4 o = f4_tanh(f4_fma(c2, k4c, f4_fma(c1, k4b, f4_mul(c0, k4a))));

        *reinterpret_cast<float4*>(out + (size_t)b * E_DIM + e) = o;

        buf ^= 1;
    }
}

extern "C" void kernel_launch(void* const* d_in, const int* in_sizes, int n_in,
                              void* d_out, int out_size, void* d_ws, size_t ws_size,
                              hipStream_t stream) {
    (void)in_sizes; (void)n_in; (void)d_ws; (void)ws_size; (void)out_size;

    const float* X     = (const float*)d_in[0];
    const float* conv1 = (const float*)d_in[1];
    const float* conv2 = (const float*)d_in[2];
    const float* conv3 = (const float*)d_in[3];
    const float* conv4 = (const float*)d_in[4];
    float* out = (float*)d_out;

    dim3 grid(E_DIM / EBLK, GRID_Y);   // (4, 1024)
    dim3 block(TPB);
    CSM_conv_tanh_chain<<<grid, block, 0, stream>>>(X, conv1, conv2, conv3, conv4, out);
}